// SSIMLoss_60361470378601
// MI455X (gfx1250) — compile-verified
//
#include <hip/hip_runtime.h>

typedef __attribute__((ext_vector_type(2))) float v2f;
typedef __attribute__((ext_vector_type(4))) float v4f;
typedef __attribute__((ext_vector_type(8))) float v8f;

namespace {
constexpr int kImg = 512;
constexpr int kBatch = 64;
constexpr int kTileH = 32;                       // rows per wave-tile
constexpr int kTileW = 16;                       // cols per wave-tile
constexpr int kTilesX = kImg / kTileW;           // 32
constexpr int kTilesY = kImg / kTileH;           // 16
constexpr int kTilesPerImg = kTilesX * kTilesY;  // 512
constexpr int kTotalTiles = kBatch * kTilesPerImg;  // 32768
constexpr int kWavesPerBlock = 4;
constexpr int kBlocks = kTotalTiles / kWavesPerBlock;  // 8192
constexpr float kC1 = 1.0e-4f;   // (0.01*1.0)^2
constexpr float kC2 = 9.0e-4f;   // (0.03*1.0)^2
// per-wave LDS region layout (in floats)
constexpr int kRawRows = kTileH + 10;        // 42 halo rows
constexpr int kRawW = 28;                    // 26 halo cols padded to 28 (K pad)
constexpr int kHbtW = 44;                    // H k-dim: 42 rows + 2 zero pad
constexpr int kXS = 0;                       // 42 x 28
constexpr int kYS = kRawRows * kRawW;        // 1176
constexpr int kHBT = 2 * kRawRows * kRawW;   // 2352: 16(n) x 44(k)
constexpr int kGW = kHBT + 16 * kHbtW;       // 3056: 11 taps (+5 pad)
constexpr int kWaveSmem = 3072;
}  // namespace

__global__ __launch_bounds__(128) void ssim_wmma_main(
    const float* __restrict__ X, const float* __restrict__ Y,
    float* __restrict__ partial) {
  __shared__ __align__(16) float smem[kWavesPerBlock * kWaveSmem];
  __shared__ float bsum[kWavesPerBlock];

  const int lane = threadIdx.x & 31;
  const int wave = threadIdx.x >> 5;
  float* ws = smem + wave * kWaveSmem;
  float* xs = ws + kXS;
  float* ys = ws + kYS;
  float* hbt = ws + kHBT;
  float* gw = ws + kGW;

  const int tile = blockIdx.x * kWavesPerBlock + wave;
  const int b = tile >> 9;           // image index (512 tiles per image)
  const int tr = (tile >> 5) & 15;   // tile row (32-row tiles)
  const int tc = tile & 31;          // tile col (16-col tiles)
  const int R0 = tr * kTileH;
  const int C0 = tc * kTileW;

  // Normalized 11-tap Gaussian (sigma = 1.5), one private copy per wave.
  {
    const float gtab[11] = {0.0010284f, 0.0075988f, 0.0360008f, 0.1093607f,
                            0.2130056f, 0.2660118f, 0.2130056f, 0.1093607f,
                            0.0360008f, 0.0075988f, 0.0010284f};
    if (lane < 11) gw[lane] = gtab[lane];
  }
  // Zero the never-written H pad rows k=42,43 (otherwise 0 x NaN hazard).
  if (lane < 16) {
    *(v2f*)(hbt + lane * kHbtW + 42) = (v2f){0.0f, 0.0f};
  }

  // Stage 42x28 halo tiles of x and y (zero padded outside the image).
  const float* xb = X + b * (kImg * kImg);
  const float* yb = Y + b * (kImg * kImg);
  for (int i = lane; i < kRawRows * kRawW; i += 32) {
    const int r = i / kRawW;
    const int c = i - r * kRawW;
    const int gr = R0 - 5 + r;
    const int gc = C0 - 5 + c;
    const bool ok = (gr >= 0) && (gr < kImg) && (gc >= 0) && (gc < kImg);
    const int off = gr * kImg + gc;
    xs[i] = ok ? xb[off] : 0.0f;
    ys[i] = ok ? yb[off] : 0.0f;
  }

  const int low = lane & 15;
  const int hi = lane >> 4;

  // Banded Gaussian fragments g[k - j]: identical per-lane values serve as B
  // in the horizontal pass and as A in the vertical pass (mirror layouts).
  v2f bandf[7];
#pragma unroll
  for (int kk = 0; kk < 7; ++kk) {
#pragma unroll
    for (int v = 0; v < 2; ++v) {
      const int k = 4 * kk + 2 * hi + v;
      const int d = k - low;
      bandf[kk][v] = (d >= 0 && d < 11) ? gw[d] : 0.0f;
    }
  }

  v8f acc0[5];  // vertical results, output rows R0+0..15
  v8f acc1[5];  // vertical results, output rows R0+16..31
#pragma unroll
  for (int ch = 0; ch < 5; ++ch) {
    // ---- horizontal pass: 3 row groups covering H rows 0..41 ----
    const int rgs[3] = {0, 12, 26};
#pragma unroll
    for (int g = 0; g < 3; ++g) {
      const int rg = rgs[g];
      const int row = rg + low;
      const float* rowx = xs + row * kRawW;
      const float* rowy = ys + row * kRawW;
      v8f hacc = {};
#pragma unroll
      for (int kk = 0; kk < 7; ++kk) {
        const int cb = 4 * kk + 2 * hi;
        v2f a;
        if (ch == 0) {
          a = *(const v2f*)(rowx + cb);
        } else if (ch == 1) {
          a = *(const v2f*)(rowy + cb);
        } else if (ch == 2) {
          const v2f ax = *(const v2f*)(rowx + cb);
          a = ax * ax;
        } else if (ch == 3) {
          const v2f ay = *(const v2f*)(rowy + cb);
          a = ay * ay;
        } else {
          const v2f ax = *(const v2f*)(rowx + cb);
          const v2f ay = *(const v2f*)(rowy + cb);
          a = ax * ay;
        }
        hacc = __builtin_amdgcn_wmma_f32_16x16x4_f32(
            false, a, false, bandf[kk], (short)0, hacc, false, false);
      }
      // Store H transposed: hbt[n=low][k = rg + 8*hi + v].
      float* dst = hbt + low * kHbtW + rg + 8 * hi;
      if ((rg & 3) == 0) {  // 16B aligned groups (rg = 0, 12)
        *(v4f*)(dst + 0) = __builtin_shufflevector(hacc, hacc, 0, 1, 2, 3);
        *(v4f*)(dst + 4) = __builtin_shufflevector(hacc, hacc, 4, 5, 6, 7);
      } else {              // rg = 26: 8B aligned only
        *(v2f*)(dst + 0) = __builtin_shufflevector(hacc, hacc, 0, 1);
        *(v2f*)(dst + 2) = __builtin_shufflevector(hacc, hacc, 2, 3);
        *(v2f*)(dst + 4) = __builtin_shufflevector(hacc, hacc, 4, 5);
        *(v2f*)(dst + 6) = __builtin_shufflevector(hacc, hacc, 6, 7);
      }
    }
    // ---- vertical pass: two independent 16-row output groups ----
#pragma unroll
    for (int vg = 0; vg < 2; ++vg) {
      v8f vacc = {};
#pragma unroll
      for (int kk = 0; kk < 7; ++kk) {
        const v2f bvec =
            *(const v2f*)(hbt + low * kHbtW + 16 * vg + 4 * kk + 2 * hi);
        vacc = __builtin_amdgcn_wmma_f32_16x16x4_f32(
            false, bandf[kk], false, bvec, (short)0, vacc, false, false);
      }
      if (vg == 0) acc0[ch] = vacc; else acc1[ch] = vacc;
    }
  }

  // Elementwise SSIM map on the 16 C-layout elements per lane.
  float ssum = 0.0f;
#pragma unroll
  for (int half = 0; half < 2; ++half) {
    const v8f* acc = half ? acc1 : acc0;
#pragma unroll
    for (int i = 0; i < 8; ++i) {
      const float mux = acc[0][i];
      const float muy = acc[1][i];
      const float exx = acc[2][i];
      const float eyy = acc[3][i];
      const float exy = acc[4][i];
      const float mux2 = mux * mux;
      const float muy2 = muy * muy;
      const float muxy = mux * muy;
      const float sx = exx - mux2;
      const float sy = eyy - muy2;
      const float sxy = exy - muxy;
      const float num = (2.0f * muxy + kC1) * (2.0f * sxy + kC2);
      const float den = (mux2 + muy2 + kC1) * (sx + sy + kC2);
      ssum += num / den;
    }
  }
  // Deterministic wave32 reduction.
#pragma unroll
  for (int off = 16; off > 0; off >>= 1) ssum += __shfl_xor(ssum, off, 32);

  if (lane == 0) bsum[wave] = ssum;
  __syncthreads();
  if (threadIdx.x == 0) {
    float t = 0.0f;
#pragma unroll
    for (int w = 0; w < kWavesPerBlock; ++w) t += bsum[w];
    partial[blockIdx.x] = t;
  }
}

__global__ __launch_bounds__(256) void ssim_finalize(
    const float* __restrict__ partial, float* __restrict__ out) {
  __shared__ double red[256];
  double s = 0.0;
  for (int i = threadIdx.x; i < kBlocks; i += 256) s += (double)partial[i];
  red[threadIdx.x] = s;
  __syncthreads();
  for (int st = 128; st > 0; st >>= 1) {
    if ((int)threadIdx.x < st) red[threadIdx.x] += red[threadIdx.x + st];
    __syncthreads();
  }
  if (threadIdx.x == 0) {
    const double npix = (double)kBatch * (double)kImg * (double)kImg;
    out[0] = (float)(1.0 - red[0] / npix);
  }
}

extern "C" void kernel_launch(void* const* d_in, const int* in_sizes, int n_in,
                              void* d_out, int out_size, void* d_ws,
                              size_t ws_size, hipStream_t stream) {
  (void)in_sizes; (void)n_in; (void)out_size; (void)ws_size;
  const float* x = (const float*)d_in[0];
  const float* y = (const float*)d_in[1];
  float* partial = (float*)d_ws;  // kBlocks floats, fully rewritten each call
  ssim_wmma_main<<<kBlocks, 128, 0, stream>>>(x, y, partial);
  ssim_finalize<<<1, 256, 0, stream>>>(partial, (float*)d_out);
}